// JTSMOutputLayers_7499012898932
// MI455X (gfx1250) — compile-verified
//
#include <hip/hip_runtime.h>
#include <math.h>

typedef __attribute__((ext_vector_type(2))) float v2f;
typedef __attribute__((ext_vector_type(4))) float v4f;
typedef __attribute__((ext_vector_type(8))) float v8f;

#define D_DIM 1024
#define C_DIM 20

__device__ __forceinline__ v8f wmma4(v2f a, v2f b, v8f c) {
    return __builtin_amdgcn_wmma_f32_16x16x4_f32(false, a, false, b, (short)0, c,
                                                 false, false);
}

// ---------------------------------------------------------------------------
// Kernel 0: init per-class stats (max=-inf, sum=0) and zero-fill the padding
// row used by out-of-range B lanes in the WMMA GEMM (cols 20..31).
// ---------------------------------------------------------------------------
__global__ __launch_bounds__(256)
void init_stats_kernel(float* __restrict__ mx, float* __restrict__ sm,
                       float* __restrict__ zpad) {
    int t = threadIdx.x;
    if (t < C_DIM) { mx[t] = -INFINITY; sm[t] = 0.0f; }
    #pragma unroll
    for (int i = 0; i < (D_DIM + 16 + 255) / 256; ++i) {
        int idx = t + i * 256;
        if (idx < D_DIM + 16) zpad[idx] = 0.0f;
    }
}

// ---------------------------------------------------------------------------
// Kernel 1: dual-head GEMM via V_WMMA_F32_16X16X4_F32.
// One wave = 32 rows (two 16-row M-tiles), software-pipelined:
// prefetch k+4 fragments before issuing the 8 WMMAs for k, so each WMMA
// waits on loads issued one iteration earlier.
//   A 16x4 f32 layout: lanes 0-15 hold K=0,1 ; lanes 16-31 hold K=2,3
//   C/D layout: VGPR j -> row j + 8*(lane>=16), col = lane&15
// Padding lanes (col 20..31) read B from a zero-filled scratch row so the
// loop is pure load+wmma and EXEC stays all-1s.
// ---------------------------------------------------------------------------
__global__ __launch_bounds__(32)
void gemm_heads_kernel(const float* __restrict__ x,
                       const float* __restrict__ Wc, const float* __restrict__ bc,
                       const float* __restrict__ Wd, const float* __restrict__ bd,
                       const float* __restrict__ zpad,
                       float* __restrict__ det_out, float* __restrict__ pcls_out,
                       float* __restrict__ cls_max)
{
    __shared__ float lds[32][64];          // cols [0,20): cls logits, [32,52): det logits

    const int r0   = blockIdx.x * 32;
    const int lane = threadIdx.x;
    const int mlo  = lane & 15;
    const int khi  = lane >> 4;            // 0: K=0,1  1: K=2,3

    v8f ac[2][2] = {{{}, {}}, {{}, {}}};   // [m-tile][n-tile] cls
    v8f ad[2][2] = {{{}, {}}, {{}, {}}};   // [m-tile][n-tile] det

    const float* xp0 = x + (size_t)(r0 + mlo) * D_DIM + 2 * khi;
    const float* xp1 = xp0 + (size_t)16 * D_DIM;

    const int n0 = mlo;                    // N-tile 0: cols 0-15
    const int n1 = 16 + mlo;               // N-tile 1: cols 16-31 (valid < 20)
    const bool n1ok = (n1 < C_DIM);

    const float* wc0 = Wc + (size_t)n0 * D_DIM + 2 * khi;
    const float* wd0 = Wd + (size_t)n0 * D_DIM + 2 * khi;
    const float* wc1 = (n1ok ? (Wc + (size_t)n1 * D_DIM) : zpad) + 2 * khi;
    const float* wd1 = (n1ok ? (Wd + (size_t)n1 * D_DIM) : zpad) + 2 * khi;

    // Pipeline prologue: fragments for k = 0.
    v2f a0 = *(const v2f*)(xp0);
    v2f a1 = *(const v2f*)(xp1);
    v2f b0 = *(const v2f*)(wc0);
    v2f b1 = *(const v2f*)(wc1);
    v2f b2 = *(const v2f*)(wd0);
    v2f b3 = *(const v2f*)(wd1);

    #pragma unroll 4
    for (int k = 0; k < D_DIM - 4; k += 4) {
        // Prefetch next iteration's fragments first (hide latency behind WMMAs).
        v2f a0n = *(const v2f*)(xp0 + k + 4);
        v2f a1n = *(const v2f*)(xp1 + k + 4);
        v2f b0n = *(const v2f*)(wc0 + k + 4);
        v2f b1n = *(const v2f*)(wc1 + k + 4);
        v2f b2n = *(const v2f*)(wd0 + k + 4);
        v2f b3n = *(const v2f*)(wd1 + k + 4);

        ac[0][0] = wmma4(a0, b0, ac[0][0]);
        ac[0][1] = wmma4(a0, b1, ac[0][1]);
        ad[0][0] = wmma4(a0, b2, ad[0][0]);
        ad[0][1] = wmma4(a0, b3, ad[0][1]);
        ac[1][0] = wmma4(a1, b0, ac[1][0]);
        ac[1][1] = wmma4(a1, b1, ac[1][1]);
        ad[1][0] = wmma4(a1, b2, ad[1][0]);
        ad[1][1] = wmma4(a1, b3, ad[1][1]);

        a0 = a0n; a1 = a1n; b0 = b0n; b1 = b1n; b2 = b2n; b3 = b3n;
    }
    // Pipeline epilogue: last k-step (k = D_DIM-4).
    ac[0][0] = wmma4(a0, b0, ac[0][0]);
    ac[0][1] = wmma4(a0, b1, ac[0][1]);
    ad[0][0] = wmma4(a0, b2, ad[0][0]);
    ad[0][1] = wmma4(a0, b3, ad[0][1]);
    ac[1][0] = wmma4(a1, b0, ac[1][0]);
    ac[1][1] = wmma4(a1, b1, ac[1][1]);
    ad[1][0] = wmma4(a1, b2, ad[1][0]);
    ad[1][1] = wmma4(a1, b3, ad[1][1]);

    // Spill logits (+bias) to LDS using the C/D VGPR layout.
    float bcv0 = bc[n0];
    float bdv0 = bd[n0];
    float bcv1 = n1ok ? bc[n1] : 0.0f;
    float bdv1 = n1ok ? bd[n1] : 0.0f;

    #pragma unroll
    for (int t = 0; t < 2; ++t) {
        #pragma unroll
        for (int j = 0; j < 8; ++j) {
            int row = t * 16 + j + 8 * khi;
            lds[row][n0]      = ac[t][0][j] + bcv0;
            lds[row][32 + n0] = ad[t][0][j] + bdv0;
            if (n1ok) {
                lds[row][n1]      = ac[t][1][j] + bcv1;
                lds[row][32 + n1] = ad[t][1][j] + bdv1;
            }
        }
    }
    __syncthreads();

    // Each lane owns one of the 32 rows: row softmax (pred_cls) + det spill.
    {
        int row = lane;
        float m = -INFINITY;
        #pragma unroll
        for (int c = 0; c < C_DIM; ++c) m = fmaxf(m, lds[row][c]);
        float e[C_DIM];
        float s = 0.0f;
        #pragma unroll
        for (int c = 0; c < C_DIM; ++c) { e[c] = __expf(lds[row][c] - m); s += e[c]; }
        float inv = 1.0f / s;

        size_t base = (size_t)(r0 + row) * C_DIM;   // 80B multiple -> 16B aligned
        #pragma unroll
        for (int q = 0; q < C_DIM / 4; ++q) {
            v4f pv;
            pv.x = e[4 * q + 0] * inv;
            pv.y = e[4 * q + 1] * inv;
            pv.z = e[4 * q + 2] * inv;
            pv.w = e[4 * q + 3] * inv;
            *(v4f*)(pcls_out + base + 4 * q) = pv;
            v4f dv = *(const v4f*)&lds[row][32 + 4 * q];
            *(v4f*)(det_out + base + 4 * q) = dv;
        }
    }
    __syncthreads();

    // Per-class tile max -> global running max (global_atomic_max_num_f32).
    if (lane < C_DIM) {
        float m = -INFINITY;
        #pragma unroll
        for (int r = 0; r < 32; ++r) m = fmaxf(m, lds[r][32 + lane]);
        atomicMax(cls_max + lane, m);
    }
}

// ---------------------------------------------------------------------------
// Kernel 2: per-class sum of exp(det - max) over all R proposals.
// ---------------------------------------------------------------------------
__global__ __launch_bounds__(256)
void det_expsum_kernel(const float* __restrict__ det, const float* __restrict__ cls_max,
                       float* __restrict__ cls_sum, int R)
{
    __shared__ float smax[C_DIM];
    __shared__ float ssum[C_DIM];
    int t = threadIdx.x;
    if (t < C_DIM) { smax[t] = cls_max[t]; ssum[t] = 0.0f; }
    __syncthreads();

    int r = blockIdx.x * blockDim.x + t;
    if (r < R) {
        const v4f* row = (const v4f*)(det + (size_t)r * C_DIM);
        v4f d[C_DIM / 4];
        #pragma unroll
        for (int q = 0; q < C_DIM / 4; ++q) d[q] = row[q];
        #pragma unroll
        for (int c = 0; c < C_DIM; ++c)
            atomicAdd(&ssum[c], __expf(d[c >> 2][c & 3] - smax[c]));  // ds_add_f32
    }
    __syncthreads();
    if (t < C_DIM) atomicAdd(cls_sum + t, ssum[t]);                   // global fp32 atomic
}

// ---------------------------------------------------------------------------
// Kernel 3: scores = pred_cls * exp(det - max) / sum
// ---------------------------------------------------------------------------
__global__ __launch_bounds__(256)
void finalize_kernel(const float* __restrict__ pcls, const float* __restrict__ det,
                     const float* __restrict__ cls_max, const float* __restrict__ cls_sum,
                     float* __restrict__ out, int R)
{
    __shared__ float smax[C_DIM];
    __shared__ float sinv[C_DIM];
    int t = threadIdx.x;
    if (t < C_DIM) { smax[t] = cls_max[t]; sinv[t] = 1.0f / cls_sum[t]; }
    __syncthreads();

    int r = blockIdx.x * blockDim.x + t;
    if (r < R) {
        size_t base = (size_t)r * C_DIM;
        const v4f* prow = (const v4f*)(pcls + base);
        const v4f* drow = (const v4f*)(det + base);
        #pragma unroll
        for (int q = 0; q < C_DIM / 4; ++q) {
            v4f p = prow[q];
            v4f d = drow[q];
            v4f o;
            #pragma unroll
            for (int j = 0; j < 4; ++j) {
                int c = 4 * q + j;
                o[j] = p[j] * __expf(d[j] - smax[c]) * sinv[c];
            }
            *(v4f*)(out + base + 4 * q) = o;
        }
    }
}

// ---------------------------------------------------------------------------
extern "C" void kernel_launch(void* const* d_in, const int* in_sizes, int n_in,
                              void* d_out, int out_size, void* d_ws, size_t ws_size,
                              hipStream_t stream) {
    const float* x  = (const float*)d_in[0];
    const float* Wc = (const float*)d_in[1];
    const float* bc = (const float*)d_in[2];
    const float* Wd = (const float*)d_in[3];
    const float* bd = (const float*)d_in[4];
    float* out = (float*)d_out;

    const int R = in_sizes[0] / D_DIM;

    // Workspace: det logits [R*C] | pred_cls [R*C] | max [C] | sum [C] | zpad [D+16]
    float* ws_det  = (float*)d_ws;
    float* ws_pcls = ws_det  + (size_t)R * C_DIM;
    float* ws_max  = ws_pcls + (size_t)R * C_DIM;
    float* ws_sum  = ws_max  + C_DIM;
    float* ws_zpad = ws_sum  + C_DIM + 24;   // alignment headroom

    init_stats_kernel<<<1, 256, 0, stream>>>(ws_max, ws_sum, ws_zpad);
    gemm_heads_kernel<<<R / 32, 32, 0, stream>>>(x, Wc, bc, Wd, bd, ws_zpad,
                                                 ws_det, ws_pcls, ws_max);
    det_expsum_kernel<<<(R + 255) / 256, 256, 0, stream>>>(ws_det, ws_max, ws_sum, R);
    finalize_kernel<<<(R + 255) / 256, 256, 0, stream>>>(ws_pcls, ws_det, ws_max, ws_sum,
                                                         out, R);
}